// GraphAttention_56788057588584
// MI455X (gfx1250) — compile-verified
//
#include <hip/hip_runtime.h>

typedef float v2f __attribute__((ext_vector_type(2)));
typedef float v8f __attribute__((ext_vector_type(8)));

#define N_HEADS   8
#define OUT_DIM   16
#define IN_DIM    128
#define HD        128   // N_HEADS * OUT_DIM
#define LEAKY     0.2f
#define NEG_INF_MAPPED 0x007FFFFFu   // ordered-uint encoding of -inf

// ---------------------------------------------------------------------------
// Kernel 0: zero d_out, init segmax (mapped -inf) and denom (0)
// ---------------------------------------------------------------------------
__global__ void gat_init(float* __restrict__ out, unsigned* __restrict__ segmax,
                         float* __restrict__ denom, int n_nodes) {
    int i = blockIdx.x * blockDim.x + threadIdx.x;
    int total = n_nodes * HD;
    if (i < total) out[i] = 0.0f;
    if (i < n_nodes * N_HEADS) {
        segmax[i] = NEG_INF_MAPPED;
        denom[i]  = 0.0f;
    }
}

// ---------------------------------------------------------------------------
// Kernel 1: ft = x @ W^T via V_WMMA_F32_16X16X4_F32.
// One wave32 per 16x16 output tile; K=128 consumed 4 at a time.
// A (16x4 f32) layout: lanes 0-15 = rows, half0 -> K{0,1}, half1 -> K{2,3}.
// B (4x16 f32) mirrors with lanes as columns. C/D: VGPR v holds row v+half*8.
// ft is row-padded to a multiple of 16 -> stores are unconditional
// (pad rows hold garbage and are never read downstream).
// ---------------------------------------------------------------------------
__global__ void gat_gemm_wmma(const float* __restrict__ x,
                              const float* __restrict__ W,
                              float* __restrict__ ft, int n_nodes) {
    const int wave = threadIdx.x >> 5;
    const int lane = threadIdx.x & 31;
    const int tiles_n = HD / 16;                    // 8 column tiles
    const int tile_id = blockIdx.x * (blockDim.x >> 5) + wave;
    const int tm = tile_id / tiles_n;
    const int tn = tile_id % tiles_n;
    if (tm * 16 >= n_nodes) return;                 // wave-uniform

    const int half = lane >> 4;                     // 0: lanes 0-15, 1: 16-31
    const int l15  = lane & 15;

    int arow = tm * 16 + l15;                       // node row (A)
    if (arow >= n_nodes) arow = n_nodes - 1;        // clamp loads only
    const int bcol = tn * 16 + l15;                 // output channel (B col)

    const float* __restrict__ xr = x + (size_t)arow * IN_DIM + half * 2;
    const float* __restrict__ wr = W + (size_t)bcol * IN_DIM + half * 2;

    v8f acc = {};
    #pragma unroll 4
    for (int k0 = 0; k0 < IN_DIM; k0 += 4) {
        v2f a, b;
        a.x = xr[k0];
        a.y = xr[k0 + 1];
        // B[k][c] = W[c][k]  (ft = x @ W^T)
        b.x = wr[k0];
        b.y = wr[k0 + 1];
        acc = __builtin_amdgcn_wmma_f32_16x16x4_f32(
                  false, a, false, b, (short)0, acc, false, false);
    }

    // Unconditional stores into padded ft: base + v*HD rows (imm offsets).
    float* __restrict__ orow = ft + (size_t)(tm * 16 + half * 8) * HD + bcol;
    #pragma unroll
    for (int v = 0; v < 8; ++v)
        orow[(size_t)v * HD] = acc[v];
}

// ---------------------------------------------------------------------------
// Kernel 2: per-node logits a1[n,h] = <ft[n,h,:], attn_l[h,:]>, same for a2
// ---------------------------------------------------------------------------
__global__ void gat_logits(const float* __restrict__ ft,
                           const float* __restrict__ attn_l,
                           const float* __restrict__ attn_r,
                           float* __restrict__ a1, float* __restrict__ a2,
                           int n_nodes) {
    int idx = blockIdx.x * blockDim.x + threadIdx.x;
    if (idx >= n_nodes * N_HEADS) return;
    const int node = idx >> 3, h = idx & 7;
    const float* f = ft + (size_t)node * HD + h * OUT_DIM;
    float s1 = 0.0f, s2 = 0.0f;
    #pragma unroll
    for (int d = 0; d < OUT_DIM; ++d) {
        s1 += f[d] * attn_l[h * OUT_DIM + d];
        s2 += f[d] * attn_r[h * OUT_DIM + d];
    }
    a1[idx] = s1;
    a2[idx] = s2;
}

__device__ __forceinline__ float edge_score(const float* a1, const float* a2,
                                            int s, int d, int h) {
    float v = a1[s * N_HEADS + h] + a2[d * N_HEADS + h];
    return v > 0.0f ? v : LEAKY * v;
}

// ---------------------------------------------------------------------------
// Kernel 3: segment-max via order-preserving uint atomics
// ---------------------------------------------------------------------------
__global__ void gat_edge_max(const float* __restrict__ a1,
                             const float* __restrict__ a2,
                             const int* __restrict__ src,
                             const int* __restrict__ dst,
                             unsigned* __restrict__ segmax, int n_edges) {
    int idx = blockIdx.x * blockDim.x + threadIdx.x;
    if (idx >= n_edges * N_HEADS) return;
    const int e = idx >> 3, h = idx & 7;
    const int s = src[e], d = dst[e];
    const float v = edge_score(a1, a2, s, d, h);
    const unsigned bits = __float_as_uint(v);
    const unsigned m = (bits & 0x80000000u) ? ~bits : (bits | 0x80000000u);
    atomicMax(&segmax[d * N_HEADS + h], m);
}

// ---------------------------------------------------------------------------
// Kernel 4: unmap segmax to float; isolated nodes (-inf) -> 0
// ---------------------------------------------------------------------------
__global__ void gat_seg_fix(unsigned* __restrict__ segmax, int n_nodes) {
    int i = blockIdx.x * blockDim.x + threadIdx.x;
    if (i >= n_nodes * N_HEADS) return;
    const unsigned m = segmax[i];
    float f;
    if (m == NEG_INF_MAPPED) {
        f = 0.0f;                                   // isolated node
    } else {
        const unsigned bits = (m & 0x80000000u) ? (m ^ 0x80000000u) : ~m;
        f = __uint_as_float(bits);
    }
    segmax[i] = __float_as_uint(f);                 // store as plain float now
}

// ---------------------------------------------------------------------------
// Kernel 5: denom[d,h] += exp(e - segmax[d,h])
// ---------------------------------------------------------------------------
__global__ void gat_edge_den(const float* __restrict__ a1,
                             const float* __restrict__ a2,
                             const float* __restrict__ segmax,
                             const int* __restrict__ src,
                             const int* __restrict__ dst,
                             float* __restrict__ denom, int n_edges) {
    int idx = blockIdx.x * blockDim.x + threadIdx.x;
    if (idx >= n_edges * N_HEADS) return;
    const int e = idx >> 3, h = idx & 7;
    const int s = src[e], d = dst[e];
    const float v  = edge_score(a1, a2, s, d, h);
    const float ee = __expf(v - segmax[d * N_HEADS + h]);
    atomicAdd(&denom[d * N_HEADS + h], ee);
}

// ---------------------------------------------------------------------------
// Kernel 6: one wave per edge. Lane l handles channels [4l,4l+4) (head l/4):
// coalesced float4 gather of ft[src] + 4 f32 atomic adds into out[dst].
// ft and out are L2-resident (25.6 MB each vs 192 MB L2).
// ---------------------------------------------------------------------------
__global__ void gat_scatter(const float* __restrict__ ft,
                            const float* __restrict__ a1,
                            const float* __restrict__ a2,
                            const float* __restrict__ segmax,
                            const float* __restrict__ denom,
                            const int* __restrict__ src,
                            const int* __restrict__ dst,
                            float* __restrict__ out, int n_edges) {
    const int gid  = blockIdx.x * blockDim.x + threadIdx.x;
    const int e    = gid >> 5;
    const int lane = threadIdx.x & 31;
    if (e >= n_edges) return;
    const int s = src[e], d = dst[e];
    const int h  = lane >> 2;                       // 4 lanes per head
    const int c0 = lane * 4;                        // channel base 0..124
    const float v     = edge_score(a1, a2, s, d, h);
    const float alpha = __expf(v - segmax[d * N_HEADS + h])
                      / denom[d * N_HEADS + h];
    const float4 f = *(const float4*)(ft + (size_t)s * HD + c0);
    float* o = out + (size_t)d * HD + c0;
    atomicAdd(o + 0, f.x * alpha);
    atomicAdd(o + 1, f.y * alpha);
    atomicAdd(o + 2, f.z * alpha);
    atomicAdd(o + 3, f.w * alpha);
}

// ---------------------------------------------------------------------------
extern "C" void kernel_launch(void* const* d_in, const int* in_sizes, int n_in,
                              void* d_out, int out_size, void* d_ws, size_t ws_size,
                              hipStream_t stream) {
    const float* x      = (const float*)d_in[0];
    const float* W      = (const float*)d_in[1];
    const float* attn_l = (const float*)d_in[2];
    const float* attn_r = (const float*)d_in[3];
    const int*   src    = (const int*)d_in[4];
    const int*   dst    = (const int*)d_in[5];
    const int n_nodes = in_sizes[0] / IN_DIM;
    const int n_edges = in_sizes[4];
    const int tiles_m = (n_nodes + 15) / 16;
    const int n_pad   = tiles_m * 16;               // padded rows for ft

    // workspace layout (floats): ft(padded) | a1 | a2 | segmax | denom  (~32 MB)
    float* ws     = (float*)d_ws;
    float* ft     = ws;
    float* a1     = ft + (size_t)n_pad * HD;
    float* a2     = a1 + (size_t)n_nodes * N_HEADS;
    unsigned* segmax = (unsigned*)(a2 + (size_t)n_nodes * N_HEADS);
    float* denom  = (float*)(segmax + (size_t)n_nodes * N_HEADS);
    float* out    = (float*)d_out;

    const int B = 256;

    // init out / segmax / denom
    {
        int total = n_nodes * HD;
        gat_init<<<(total + B - 1) / B, B, 0, stream>>>(out, segmax, denom, n_nodes);
    }
    // WMMA GEMM: one wave per 16x16 tile, 8 waves per block
    {
        const int tiles = tiles_m * (HD / 16);
        gat_gemm_wmma<<<(tiles + 7) / 8, B, 0, stream>>>(x, W, ft, n_nodes);
    }
    // per-node logits
    {
        int total = n_nodes * N_HEADS;
        gat_logits<<<(total + B - 1) / B, B, 0, stream>>>(ft, attn_l, attn_r, a1, a2, n_nodes);
    }
    // segment max
    {
        int total = n_edges * N_HEADS;
        gat_edge_max<<<(total + B - 1) / B, B, 0, stream>>>(a1, a2, src, dst, segmax, n_edges);
    }
    // unmap + isolated-node fix
    {
        int total = n_nodes * N_HEADS;
        gat_seg_fix<<<(total + B - 1) / B, B, 0, stream>>>(segmax, n_nodes);
    }
    // segment sum of exp
    {
        int total = n_edges * N_HEADS;
        gat_edge_den<<<(total + B - 1) / B, B, 0, stream>>>(
            a1, a2, (const float*)segmax, src, dst, denom, n_edges);
    }
    // weighted scatter: one wave per edge
    {
        long long threads = (long long)n_edges * 32;
        gat_scatter<<<(int)((threads + B - 1) / B), B, 0, stream>>>(
            ft, a1, a2, (const float*)segmax, denom, src, dst, out, n_edges);
    }
}